// Conditional_InstanceNorm2d_45329084841999
// MI455X (gfx1250) — compile-verified
//
#include <hip/hip_runtime.h>
#include <hip/hip_bf16.h>

#define IN_EPS 1e-5f

typedef float f32x4 __attribute__((ext_vector_type(4)));
typedef int   v4i   __attribute__((vector_size(16)));   // type the async builtin expects

#if defined(__gfx1250__) && __has_builtin(__builtin_amdgcn_global_load_async_to_lds_b128)
#define HAVE_ASYNC_LDS 1
typedef __attribute__((address_space(1))) v4i*  g_v4i_ptr;   // global (device) int4*
typedef __attribute__((address_space(3))) v4i*  l_v4i_ptr;   // LDS int4*
#else
#define HAVE_ASYNC_LDS 0
#endif

// Fixed shapes from the reference model (compile-time so address math is
// shifts/masks and all loops fully unroll).
constexpr int kC   = 256;
constexpr int kHW  = 64 * 64;          // 4096 floats = 16 KB tile
constexpr int kThr = 256;              // 8 wave32
constexpr int kHW4 = kHW / 4;          // 1024 float4s
constexpr int kIter = kHW4 / kThr;     // 4 float4s per thread

// One workgroup per (b,c) channel-instance.
// Stage the 16 KB tile in LDS via gfx1250 async global->LDS DMA (ASYNCcnt),
// single pass for sum/sumsq, wave32 butterfly + cross-wave LDS reduce,
// then normalize from LDS and stream out with nontemporal b128 stores.
// x is read from HBM exactly once; out written exactly once.
__global__ __launch_bounds__(kThr) void cond_in2d_kernel(
    const float* __restrict__ x,
    const int* __restrict__ style_id,
    const float* __restrict__ gamma,
    const float* __restrict__ beta,
    float* __restrict__ out)
{
    __shared__ float smem[kHW + 64];           // tile + reduction scratch
    float* red = smem + kHW;                   // red[0..31]=sum, red[32..63]=sumsq

    const int bc  = blockIdx.x;
    const int b   = bc >> 8;                   // bc / kC
    const int c   = bc & (kC - 1);
    const int tid = threadIdx.x;

    const f32x4* __restrict__ x4 = (const f32x4*)(x + (size_t)bc * kHW);
    f32x4* __restrict__ o4       = (f32x4*)(out + (size_t)bc * kHW);
    f32x4* s4                    = (f32x4*)smem;

    // ---- Stage tile: global -> LDS (async DMA) -------------------------
#if HAVE_ASYNC_LDS
    {
        __attribute__((address_space(1))) char* gp =
            (__attribute__((address_space(1))) char*)x4;
        __attribute__((address_space(3))) char* lp =
            (__attribute__((address_space(3))) char*)s4;
#pragma unroll
        for (int r = 0; r < kIter; ++r) {
            const int i4 = r * kThr + tid;
            __builtin_amdgcn_global_load_async_to_lds_b128(
                (g_v4i_ptr)(gp + (size_t)i4 * 16),
                (l_v4i_ptr)(lp + (size_t)i4 * 16),
                0, 0);
        }
#if __has_builtin(__builtin_amdgcn_s_wait_asynccnt)
        __builtin_amdgcn_s_wait_asynccnt(0);
#else
        asm volatile("s_wait_asynccnt 0" ::: "memory");
#endif
    }
#else
#pragma unroll
    for (int r = 0; r < kIter; ++r) {
        const int i4 = r * kThr + tid;
        s4[i4] = x4[i4];
    }
#endif
    __syncthreads();

    // ---- Pass 1: sum and sum of squares from LDS -----------------------
    float sum = 0.0f, sq = 0.0f;
#pragma unroll
    for (int r = 0; r < kIter; ++r) {
        f32x4 v = s4[r * kThr + tid];
        sum += (v.x + v.y) + (v.z + v.w);
        sq  += (v.x * v.x + v.y * v.y) + (v.z * v.z + v.w * v.w);
    }
    // wave32 butterfly reduction
#pragma unroll
    for (int off = 16; off > 0; off >>= 1) {
        sum += __shfl_xor(sum, off, 32);
        sq  += __shfl_xor(sq,  off, 32);
    }
    const int wave = tid >> 5;
    if ((tid & 31) == 0) { red[wave] = sum; red[32 + wave] = sq; }
    __syncthreads();

    float tsum = 0.0f, tsq = 0.0f;
#pragma unroll
    for (int w = 0; w < kThr / 32; ++w) { tsum += red[w]; tsq += red[32 + w]; }

    constexpr float inv = 1.0f / (float)kHW;
    const float mean = tsum * inv;
    const float var  = tsq * inv - mean * mean;
    const float rs   = rsqrtf(var + IN_EPS);

    const int   sid   = style_id[b];
    const float g     = gamma[sid * kC + c];
    const float bt    = beta[sid * kC + c];
    const float scale = g * rs;
    const float shift = bt - mean * scale;

    // ---- Pass 2: normalize from LDS, stream out (nontemporal) ----------
#pragma unroll
    for (int r = 0; r < kIter; ++r) {
        const int i4 = r * kThr + tid;
        f32x4 v = s4[i4];
        f32x4 o;
        o.x = fmaf(v.x, scale, shift);
        o.y = fmaf(v.y, scale, shift);
        o.z = fmaf(v.z, scale, shift);
        o.w = fmaf(v.w, scale, shift);
        __builtin_nontemporal_store(o, &o4[i4]);
    }
}

extern "C" void kernel_launch(void* const* d_in, const int* in_sizes, int n_in,
                              void* d_out, int out_size, void* d_ws, size_t ws_size,
                              hipStream_t stream) {
    const float* x     = (const float*)d_in[0];
    const int*   sid   = (const int*)d_in[1];
    const float* gamma = (const float*)d_in[2];
    const float* beta  = (const float*)d_in[3];
    float*       out   = (float*)d_out;

    // One block per (b,c) pair; shapes fixed by the reference model.
    const int blocks = in_sizes[0] / kHW;       // B*C = 8192

    cond_in2d_kernel<<<dim3(blocks), dim3(kThr), 0, stream>>>(
        x, sid, gamma, beta, out);
}